// GraphUnpoolingMesh_84980222918930
// MI455X (gfx1250) — compile-verified
//
#include <hip/hip_runtime.h>
#include <hip/hip_bf16.h>

typedef float v4f __attribute__((ext_vector_type(4)));

#define FDIM 128

// ---------------------------------------------------------------------------
// Kernel 1: bulk copy x -> new_x[0:N] through the CDNA5 async global<->LDS
// DMA path (ASYNCcnt-tracked). Each block stages a 16 KB tile in LDS:
// 4 rounds x 256 lanes x 16 B. No VGPR data movement.
// ---------------------------------------------------------------------------
__global__ void __launch_bounds__(256) gum_copy_x_async(const float* __restrict__ x,
                                                        float* __restrict__ out) {
  __shared__ v4f buf[1024];  // 16 KB staging
  const unsigned t = threadIdx.x;
  const unsigned base4 = blockIdx.x * 1024u;  // float4 index of this block's tile

  unsigned goff[4], loff[4];
#pragma unroll
  for (int c = 0; c < 4; ++c) {
    const unsigned idx = base4 + (unsigned)c * 256u + t;
    goff[c] = idx * 16u;  // global byte offset from base pointer
    // low 32 bits of the flat address of an LDS location == LDS byte offset
    loff[c] = (unsigned)(size_t)&buf[c * 256 + t];
  }

#pragma unroll
  for (int c = 0; c < 4; ++c) {
    asm volatile("global_load_async_to_lds_b128 %0, %1, %2"
                 :
                 : "v"(loff[c]), "v"(goff[c]), "s"(x)
                 : "memory");
  }
  // loads complete in order; wait for all 4 before the stores read LDS
  asm volatile("s_wait_asynccnt 0" ::: "memory");

#pragma unroll
  for (int c = 0; c < 4; ++c) {
    asm volatile("global_store_async_from_lds_b128 %0, %1, %2"
                 :
                 : "v"(goff[c]), "v"(loff[c]), "s"(out)
                 : "memory");
  }
  asm volatile("s_wait_asynccnt 0" ::: "memory");
}

// Scalar tail for any remainder not covered by the 16 KB tiles.
__global__ void gum_copy_tail(const float* __restrict__ x, float* __restrict__ out,
                              long long start, long long total) {
  long long i = start + (long long)blockIdx.x * blockDim.x + threadIdx.x;
  if (i < total) out[i] = x[i];
}

// ---------------------------------------------------------------------------
// Kernel 2: midpoint rows. One wave32 per edge; lane l moves float4 #l of the
// 128-float row -> one coalesced 512 B request per row per wave. x rows stay
// resident in L2 (51 MB < 192 MB); mid rows leave via non-temporal stores so
// the 307 MB output stream does not evict x.
// ---------------------------------------------------------------------------
__global__ void __launch_bounds__(256) gum_midpoints(const float* __restrict__ x,
                                                     const long long* __restrict__ ei,
                                                     float* __restrict__ newx,
                                                     int N, int E) {
  const unsigned gid = blockIdx.x * blockDim.x + threadIdx.x;
  const unsigned e = gid >> 5;
  const unsigned lane = gid & 31u;
  if (e >= (unsigned)E) return;

  const long long s = ei[e];              // edge_index[0][e]
  const long long d = ei[(size_t)E + e];  // edge_index[1][e]

  const v4f* ra = (const v4f*)(x + (size_t)s * FDIM) + lane;
  const v4f* rb = (const v4f*)(x + (size_t)d * FDIM) + lane;
  const v4f a = *ra;
  const v4f b = *rb;
  const v4f m = (a + b) * 0.5f;

  v4f* o = (v4f*)(newx + ((size_t)N + e) * FDIM) + lane;
  __builtin_nontemporal_store(m, o);  // TH=NT: keep x resident in L2
}

// ---------------------------------------------------------------------------
// Kernel 3: new_ei rows. Output layout is [2, M] row-major (sources then
// dests), M = 3E + 3T. Regions: [0,E) original edges, [E,2E) src->mid,
// [2E,3E) mid->dst, [3E,3E+3T) triangle edges (i,i+1),(i+1,i+2),(i+2,i).
// ---------------------------------------------------------------------------
__global__ void gum_edges(const long long* __restrict__ ei,
                          float* __restrict__ srcRow, float* __restrict__ dstRow,
                          int N, int E, int T) {
  const int t = blockIdx.x * blockDim.x + threadIdx.x;
  const int M = 3 * E + 3 * T;
  if (t >= M) return;

  float s, d;
  if (t < E) {
    s = (float)ei[t];
    d = (float)ei[(size_t)E + t];
  } else if (t < 2 * E) {
    const int e = t - E;
    s = (float)ei[e];
    d = (float)(N + e);
  } else if (t < 3 * E) {
    const int e = t - 2 * E;
    s = (float)(N + e);
    d = (float)ei[(size_t)E + e];
  } else {
    const int k = t - 3 * E;
    const int q = k / 3;
    const int r = k - 3 * q;
    const int i = N + 3 * q;
    s = (float)(i + r);
    d = (float)(i + ((r + 1) % 3));
  }
  srcRow[t] = s;
  dstRow[t] = d;
}

// ---------------------------------------------------------------------------
extern "C" void kernel_launch(void* const* d_in, const int* in_sizes, int n_in,
                              void* d_out, int out_size, void* d_ws, size_t ws_size,
                              hipStream_t stream) {
  (void)n_in; (void)out_size; (void)d_ws; (void)ws_size;

  const float* x = (const float*)d_in[0];
  const long long* ei = (const long long*)d_in[1];
  float* out = (float*)d_out;

  const int N = in_sizes[0] / FDIM;
  const int E = in_sizes[1] / 2;
  const int T = (E > 2) ? (E / 3) : 0;  // == ceil((E-2)/3) for E > 2
  const int M = 3 * E + 3 * T;
  const size_t NX = (size_t)(N + E) * FDIM;  // floats in new_x

  // 1) x -> new_x[0:N] via async DMA tiles (16 KB / block), scalar tail if any
  const long long totalF = (long long)N * FDIM;
  const long long n4 = totalF / 4;
  const int blocks1 = (int)(n4 / 1024);
  if (blocks1 > 0) {
    gum_copy_x_async<<<blocks1, 256, 0, stream>>>(x, out);
  }
  const long long done = (long long)blocks1 * 4096;
  if (done < totalF) {
    const long long tail = totalF - done;
    gum_copy_tail<<<(int)((tail + 255) / 256), 256, 0, stream>>>(x, out, done, totalF);
  }

  // 2) midpoint rows: one wave32 per edge
  if (E > 0) {
    const long long thr = (long long)E * 32;
    gum_midpoints<<<(int)((thr + 255) / 256), 256, 0, stream>>>(x, ei, out, N, E);
  }

  // 3) edge index rows
  if (M > 0) {
    gum_edges<<<(M + 255) / 256, 256, 0, stream>>>(ei, out + NX, out + NX + (size_t)M,
                                                   N, E, T);
  }
}